// EdgeSampleAggregate_44367012168298
// MI455X (gfx1250) — compile-verified
//
#include <hip/hip_runtime.h>
#include <hip/hip_bf16.h>
#include <math.h>

typedef __attribute__((ext_vector_type(2))) float v2f;
typedef __attribute__((ext_vector_type(4))) float v4f;
typedef __attribute__((ext_vector_type(8))) float v8f;

// ---------------------------------------------------------------------------
// Kernel 1: gather + mean aggregation (b128 gather path).
// One block (256 threads) per edge.
//   f_u[e, 0:64]  = (1/64) * sum_{64 rows} h[idx_e]
//   f_u[e,64:128] = (1/64) * sum_{64 rows} m[idx_n]
// Rows are 64 floats = 16 float4. Thread t: quad = t&15 (16B column chunk),
// rslot = t>>4 (row slot). Each thread accumulates 4 rows with
// global_load_b128, then a stride-16 LDS tree reduces 16 row slots.
// ---------------------------------------------------------------------------
__global__ __launch_bounds__(256)
void agg_kernel(const float* __restrict__ h, const float* __restrict__ m,
                const int* __restrict__ idx_e, const int* __restrict__ idx_n,
                float* __restrict__ f_u) {
    const int e     = blockIdx.x;
    const int tid   = threadIdx.x;
    const int quad  = tid & 15;   // which float4 of the 256B row
    const int rslot = tid >> 4;   // 0..15 row slot

    __shared__ int s_ie[64];
    __shared__ int s_in[64];
    if (tid < 64)          s_ie[tid]      = idx_e[(size_t)e * 64 + tid];
    else if (tid < 128)    s_in[tid - 64] = idx_n[(size_t)e * 64 + (tid - 64)];
    __syncthreads();

    const v4f* __restrict__ hv = (const v4f*)h;   // rows of 16 float4
    const v4f* __restrict__ mv = (const v4f*)m;
    v4f* __restrict__ fv       = (v4f*)f_u;       // 32 float4 per edge

    __shared__ v4f red[256];

    // ---- h part ----
    {
        v4f acc = {0.f, 0.f, 0.f, 0.f};
        #pragma unroll
        for (int i = 0; i < 4; ++i) {
            const int r = rslot + 16 * i;
            acc += hv[(size_t)s_ie[r] * 16 + quad];
        }
        red[tid] = acc;
        __syncthreads();
        #pragma unroll
        for (int s = 128; s >= 16; s >>= 1) {
            if (tid < s) red[tid] += red[tid + s];
            __syncthreads();
        }
        if (tid < 16) {
            v4f r = red[tid];
            r *= (1.0f / 64.0f);
            fv[(size_t)e * 32 + tid] = r;          // f_u[e, 0:64]
        }
        __syncthreads();
    }

    // ---- m part ----
    {
        v4f acc = {0.f, 0.f, 0.f, 0.f};
        #pragma unroll
        for (int i = 0; i < 4; ++i) {
            const int r = rslot + 16 * i;
            acc += mv[(size_t)s_in[r] * 16 + quad];
        }
        red[tid] = acc;
        __syncthreads();
        #pragma unroll
        for (int s = 128; s >= 16; s >>= 1) {
            if (tid < s) red[tid] += red[tid + s];
            __syncthreads();
        }
        if (tid < 16) {
            v4f r = red[tid];
            r *= (1.0f / 64.0f);
            fv[(size_t)e * 32 + 16 + tid] = r;     // f_u[e, 64:128]
        }
    }
}

// ---------------------------------------------------------------------------
// Kernel 2: fp32 GEMM  C[Mx128] = A[MxKd] * B[Kdx128]  via V_WMMA_F32_16X16X4_F32.
// One wave (32 lanes) per 16x16 output tile; 8 waves per block. N fixed = 128
// (8 N-tiles), so tile decomposition is shift/mask (no SALU div).
// ISA layouts (CDNA5 ISA):
//   A 16x4: lanes 0-15 hold M=lane, VGPR0=K0/VGPR1=K1; lanes 16-31 hold
//           M=lane-16, VGPR0=K2/VGPR1=K3.
//   B 4x16: row K striped across lanes within a VGPR, same K split.
//   C/D 16x16: VGPR v, lanes 0-15 -> M=v, lanes 16-31 -> M=v+8; N=lane%16.
// Tile indices are wave-uniform, so EXEC stays all-ones around the WMMA.
// ---------------------------------------------------------------------------
__global__ __launch_bounds__(256)
void wmma_gemm_f32(const float* __restrict__ A, const float* __restrict__ B,
                   float* __restrict__ C, int M, int Kd) {
    constexpr int N = 128;
    const int wave = threadIdx.x >> 5;
    const int lane = threadIdx.x & 31;
    const int tile = blockIdx.x * 8 + wave;
    const int tm   = tile >> 3;          // row tile (N/16 == 8)
    const int tn   = tile & 7;           // col tile
    if (tm * 16 >= M) return;            // wave-uniform

    const int half = lane >> 4;          // 0: K+0/K+1, 1: K+2/K+3
    const int l    = lane & 15;
    const int arow = tm * 16 + l;        // A row for this lane
    const int ncol = tn * 16 + l;        // B/C column for this lane

    v8f acc = {0.f, 0.f, 0.f, 0.f, 0.f, 0.f, 0.f, 0.f};

    for (int k = 0; k < Kd; k += 4) {
        const int ka = k + half * 2;
        v2f a, b;
        a.x = A[(size_t)arow * Kd + ka];
        a.y = A[(size_t)arow * Kd + ka + 1];
        b.x = B[(size_t)(ka)     * N + ncol];
        b.y = B[(size_t)(ka + 1) * N + ncol];
        acc = __builtin_amdgcn_wmma_f32_16x16x4_f32(
            /*neg_a=*/false, a, /*neg_b=*/false, b,
            /*c_mod=*/(short)0, acc, /*reuse_a=*/false, /*reuse_b=*/false);
    }

    const int r0 = tm * 16 + half * 8;
    #pragma unroll
    for (int v = 0; v < 8; ++v) {
        C[(size_t)(r0 + v) * N + ncol] = acc[v];
    }
}

// ---------------------------------------------------------------------------
// Kernel 3: attention head. One block (128 threads) per edge.
// scores = [dot(tanh(ph),a), dot(tanh(pu),a), same], softmax, fused output:
//   h_new = a0*ph + (a1+a2)*pu
// ---------------------------------------------------------------------------
__global__ __launch_bounds__(128)
void attn_kernel(const float* __restrict__ ph, const float* __restrict__ pu,
                 const float* __restrict__ a, float* __restrict__ h_new,
                 float* __restrict__ alpha_out) {
    const int e = blockIdx.x;
    const int d = threadIdx.x;           // 0..127

    const float vph = ph[(size_t)e * 128 + d];
    const float vpu = pu[(size_t)e * 128 + d];
    const float ad  = a[d];

    __shared__ float r0[128];
    __shared__ float r1[128];
    r0[d] = tanhf(vph) * ad;
    r1[d] = tanhf(vpu) * ad;
    __syncthreads();

    #pragma unroll
    for (int s = 64; s > 0; s >>= 1) {
        if (d < s) { r0[d] += r0[d + s]; r1[d] += r1[d + s]; }
        __syncthreads();
    }

    __shared__ float al0, al12;
    if (d == 0) {
        const float s0 = r0[0], s1 = r1[0];        // s2 == s1
        const float mx = fmaxf(s0, s1);
        const float e0 = expf(s0 - mx);
        const float e1 = expf(s1 - mx);
        const float denom = e0 + 2.f * e1;
        const float a0 = e0 / denom;
        const float a1 = e1 / denom;
        al0  = a0;
        al12 = 2.f * a1;
        alpha_out[(size_t)e * 3 + 0] = a0;
        alpha_out[(size_t)e * 3 + 1] = a1;
        alpha_out[(size_t)e * 3 + 2] = a1;
    }
    __syncthreads();

    h_new[(size_t)e * 128 + d] = al0 * vph + al12 * vpu;
}

// ---------------------------------------------------------------------------
// Launch. Inputs (setup_inputs order):
//  0 h_xe [E,64]  1 h [NE,64]  2 m [NN,64]  3 W_h [64,128]  4 W_f [128,128]
//  5 a [128]      6 idx_edge_u [E,K,L]  7 idx_node_u  8 idx_edge_v (unused)
//  9 idx_node_v (unused)
// Output: h_xe_new [E,128] then alpha [E,3], concatenated flat (fp32).
// ---------------------------------------------------------------------------
extern "C" void kernel_launch(void* const* d_in, const int* in_sizes, int n_in,
                              void* d_out, int out_size, void* d_ws, size_t ws_size,
                              hipStream_t stream) {
    const float* h_xe = (const float*)d_in[0];
    const float* h    = (const float*)d_in[1];
    const float* m    = (const float*)d_in[2];
    const float* W_h  = (const float*)d_in[3];
    const float* W_f  = (const float*)d_in[4];
    const float* avec = (const float*)d_in[5];
    const int* idx_e  = (const int*)d_in[6];
    const int* idx_n  = (const int*)d_in[7];

    const int E  = in_sizes[0] / 64;     // 8192
    const int DH = in_sizes[5];          // 128

    float* f_u = (float*)d_ws;                       // [E,128]
    float* ph  = f_u + (size_t)E * DH;               // [E,128]
    float* pu  = ph  + (size_t)E * DH;               // [E,128]

    float* h_new     = (float*)d_out;                // [E,128]
    float* alpha_out = (float*)d_out + (size_t)E * DH;  // [E,3]

    // 1) gather + aggregate -> f_u  (b128 gather, bandwidth-dominant step)
    agg_kernel<<<E, 256, 0, stream>>>(h, m, idx_e, idx_n, f_u);

    // 2) WMMA GEMMs: ph = h_xe @ W_h (K=64), pu = f_u @ W_f (K=128)
    const int tiles  = (E / 16) * (DH / 16);         // 4096
    const int blocks = tiles / 8;                    // 8 waves per block
    wmma_gemm_f32<<<blocks, 256, 0, stream>>>(h_xe, W_h, ph, E, 64);
    wmma_gemm_f32<<<blocks, 256, 0, stream>>>(f_u,  W_f, pu, E, 128);

    // 3) attention + output
    attn_kernel<<<E, 128, 0, stream>>>(ph, pu, avec, h_new, alpha_out);
}